// GCNClassifier_30786325577983
// MI455X (gfx1250) — compile-verified
//
#include <hip/hip_runtime.h>
#include <hip/hip_bf16.h>

typedef __attribute__((ext_vector_type(2))) float v2f;
typedef __attribute__((ext_vector_type(8))) float v8f;

#define N_NODES 100000
#define FEAT1 64
#define FEAT2 32

// ---------------- zero init ----------------
__global__ void zero_kernel(float* __restrict__ p, int n) {
    int i = blockIdx.x * blockDim.x + threadIdx.x;
    if (i < n) p[i] = 0.0f;
}

// ---------------- degree: atomic count on target nodes ----------------
__global__ void deg_kernel(const long long* __restrict__ col,
                           float* __restrict__ deg, int E) {
    int e = blockIdx.x * blockDim.x + threadIdx.x;
    if (e < E) atomicAdd(&deg[(int)col[e]], 1.0f);
}

// deg -> 1/sqrt(deg + 1)   (+1 = self loop; always > 0)
__global__ void dinv_kernel(float* __restrict__ deg, int n) {
    int i = blockIdx.x * blockDim.x + threadIdx.x;
    if (i < n) deg[i] = rsqrtf(deg[i] + 1.0f);
}

// ---------------- layer-1 scalar aggregation + norm cache ----------------
// s[c] += dinv[r]*dinv[c]*x[r];  norm[e] stored for reuse
__global__ void l1_scatter(const long long* __restrict__ row,
                           const long long* __restrict__ col,
                           const float* __restrict__ x,
                           const float* __restrict__ dinv,
                           float* __restrict__ s,
                           float* __restrict__ norm, int E) {
    int e = blockIdx.x * blockDim.x + threadIdx.x;
    if (e < E) {
        int r = (int)row[e];
        int c = (int)col[e];
        float nrm = dinv[r] * dinv[c];
        norm[e] = nrm;
        atomicAdd(&s[c], nrm * x[r]);
    }
}

// h1[n][f] = relu((s[n] + dinv[n]^2*x[n]) * W1[f] + b1[f])
__global__ void h1_kernel(const float* __restrict__ s,
                          const float* __restrict__ x,
                          const float* __restrict__ dinv,
                          const float* __restrict__ W1,
                          const float* __restrict__ b1,
                          float* __restrict__ h1, int n) {
    int t = blockIdx.x * blockDim.x + threadIdx.x;
    int node = t >> 6;
    int f = t & 63;
    if (node < n) {
        float di = dinv[node];
        float sv = s[node] + di * di * x[node];
        float v = fmaf(sv, W1[f], b1[f]);
        h1[t] = v > 0.0f ? v : 0.0f;
    }
}

// ---------------- dense GEMM m = h1 @ W2 via f32 WMMA ----------------
// One wave computes a 16x32 tile of m: two 16x16 accumulators, K=64 in 16 steps.
__global__ __launch_bounds__(256) void gemm_wmma(const float* __restrict__ h1,
                                                 const float* __restrict__ W2,
                                                 float* __restrict__ m, int n) {
    int gwave = (blockIdx.x * blockDim.x + threadIdx.x) >> 5;
    int lane  = threadIdx.x & 31;
    int row0  = gwave * 16;
    if (row0 >= n) return;  // uniform per wave -> EXEC stays all-ones for WMMA

    // A-matrix 16x4 f32 layout: lanes 0-15 hold (K=0,K=1), lanes 16-31 hold (K=2,K=3)
    int mrow  = row0 + (lane & 15);
    int khalf = (lane >> 4) * 2;   // 0 or 2
    int nlo   = lane & 15;

    v8f c0 = {};
    v8f c1 = {};
#pragma unroll
    for (int k0 = 0; k0 < FEAT1; k0 += 4) {
        v2f a = *(const v2f*)(h1 + (size_t)mrow * FEAT1 + k0 + khalf);
        int kb = k0 + khalf;
        v2f b0, b1;
        b0.x = W2[kb * FEAT2 + nlo];
        b0.y = W2[(kb + 1) * FEAT2 + nlo];
        b1.x = W2[kb * FEAT2 + 16 + nlo];
        b1.y = W2[(kb + 1) * FEAT2 + 16 + nlo];
        c0 = __builtin_amdgcn_wmma_f32_16x16x4_f32(false, a, false, b0,
                                                   (short)0, c0, false, false);
        c1 = __builtin_amdgcn_wmma_f32_16x16x4_f32(false, a, false, b1,
                                                   (short)0, c1, false, false);
    }
    // C/D layout: VGPR vi -> M = vi (lanes 0-15), M = 8+vi (lanes 16-31)
    int rbase = row0 + (lane >> 4) * 8;
#pragma unroll
    for (int vi = 0; vi < 8; ++vi) {
        m[(size_t)(rbase + vi) * FEAT2 + nlo]      = c0[vi];
        m[(size_t)(rbase + vi) * FEAT2 + 16 + nlo] = c1[vi];
    }
}

// ---------------- layer-2 aggregation: wave per edge, lane per feature ----------------
__global__ void l2_scatter(const long long* __restrict__ row,
                           const long long* __restrict__ col,
                           const float* __restrict__ norm,
                           const float* __restrict__ m,
                           float* __restrict__ agg2, int E) {
    int gid = blockIdx.x * blockDim.x + threadIdx.x;
    int e = gid >> 5;
    int f = gid & 31;
    if (e < E) {
        int r = (int)row[e];     // same address across wave -> single broadcast load
        int c = (int)col[e];
        float nrm = norm[e];
        atomicAdd(&agg2[(size_t)c * FEAT2 + f], nrm * m[(size_t)r * FEAT2 + f]);
    }
}

// ---------------- fused: h2 = relu(agg2 + self*m + b2); z = h2 @ W3;
// also adds layer-3 self-loop term dinv^2*z[n] into the global accumulator ----------------
__global__ void h2z_kernel(const float* __restrict__ agg2,
                           const float* __restrict__ m,
                           const float* __restrict__ dinv,
                           const float* __restrict__ b2,
                           const float* __restrict__ W3,
                           float* __restrict__ z,
                           float* __restrict__ acc, int n) {
    int gid = blockIdx.x * blockDim.x + threadIdx.x;
    int node = gid >> 5;
    int f = gid & 31;
    if (node >= n) return;
    float di = dinv[node];
    float self = di * di;
    size_t idx = (size_t)node * FEAT2 + f;
    float v = agg2[idx] + self * m[idx] + b2[f];
    v = fmaxf(v, 0.0f);
    float p = v * W3[f];
#pragma unroll
    for (int off = 16; off > 0; off >>= 1) p += __shfl_down(p, off, 32);
    if (f == 0) {
        z[node] = p;
        atomicAdd(acc, self * p);
    }
}

// ---------------- layer-3 edge reduction: acc += sum_e norm[e]*z[row[e]] ----------------
__global__ __launch_bounds__(256) void l3_reduce(const long long* __restrict__ row,
                                                 const float* __restrict__ norm,
                                                 const float* __restrict__ z,
                                                 float* __restrict__ acc, int E) {
    __shared__ float sdata[256];
    float local = 0.0f;
    for (int e = blockIdx.x * blockDim.x + threadIdx.x; e < E;
         e += gridDim.x * blockDim.x) {
        local += norm[e] * z[(int)row[e]];
    }
    sdata[threadIdx.x] = local;
    __syncthreads();
#pragma unroll
    for (int sft = 128; sft > 0; sft >>= 1) {
        if ((int)threadIdx.x < sft) sdata[threadIdx.x] += sdata[threadIdx.x + sft];
        __syncthreads();
    }
    if (threadIdx.x == 0) atomicAdd(acc, sdata[0]);
}

// ---------------- finalize: sigmoid(acc + N*b3) ----------------
__global__ void finalize_kernel(const float* __restrict__ acc,
                                const float* __restrict__ b3,
                                float* __restrict__ out, int n) {
    float t = acc[0] + (float)n * b3[0];
    out[0] = 1.0f / (1.0f + expf(-t));
}

extern "C" void kernel_launch(void* const* d_in, const int* in_sizes, int n_in,
                              void* d_out, int out_size, void* d_ws, size_t ws_size,
                              hipStream_t stream) {
    const float* x       = (const float*)d_in[0];
    const long long* ei  = (const long long*)d_in[1];   // int64 [2, E]
    const float* W1      = (const float*)d_in[2];
    const float* b1      = (const float*)d_in[3];
    const float* W2      = (const float*)d_in[4];
    const float* b2      = (const float*)d_in[5];
    const float* W3      = (const float*)d_in[6];
    const float* b3      = (const float*)d_in[7];
    float* out           = (float*)d_out;

    const int N = in_sizes[0];          // 100000
    const int E = in_sizes[1] / 2;      // 3200000
    const long long* row = ei;
    const long long* col = ei + E;

    // workspace layout (floats)
    float* ws   = (float*)d_ws;
    float* dinv = ws;                        // N
    float* s    = ws + (size_t)N;            // N
    float* agg2 = ws + (size_t)2 * N;        // 32N
    float* z    = ws + (size_t)34 * N;       // N
    float* acc  = ws + (size_t)35 * N;       // 1 (+15 pad)
    float* h1   = ws + (size_t)35 * N + 16;  // 64N
    float* mbuf = ws + (size_t)99 * N + 16;  // 32N
    float* norm = ws + (size_t)131 * N + 16; // E

    const int B = 256;

    // 1) zero dinv, s, agg2, z, acc  (35N + 16 floats)
    int nzero = 35 * N + 16;
    zero_kernel<<<(nzero + B - 1) / B, B, 0, stream>>>(ws, nzero);

    // 2) degree count + transform to rsqrt(deg+1)
    deg_kernel<<<(E + B - 1) / B, B, 0, stream>>>(col, dinv, E);
    dinv_kernel<<<(N + B - 1) / B, B, 0, stream>>>(dinv, N);

    // 3) layer-1 scalar aggregation (rank-1 trick) + norm cache
    l1_scatter<<<(E + B - 1) / B, B, 0, stream>>>(row, col, x, dinv, s, norm, E);

    // 4) h1 = relu(s_total * W1 + b1)   [N,64]
    {
        long long t = (long long)N * FEAT1;
        h1_kernel<<<(unsigned)((t + B - 1) / B), B, 0, stream>>>(s, x, dinv, W1, b1, h1, N);
    }

    // 5) m = h1 @ W2 via f32 WMMA (wave per 16x32 tile)
    {
        int waves = (N + 15) / 16;
        int threads = waves * 32;
        gemm_wmma<<<(threads + B - 1) / B, B, 0, stream>>>(h1, W2, mbuf, N);
    }

    // 6) layer-2 aggregation: wave per edge, 32 lanes = 32 features
    {
        long long t = (long long)E * 32;
        l2_scatter<<<(unsigned)((t + B - 1) / B), B, 0, stream>>>(row, col, norm, mbuf, agg2, E);
    }

    // 7) fused h2 + z = h2@W3 + layer-3 self-loop accumulation
    {
        long long t = (long long)N * 32;
        h2z_kernel<<<(unsigned)((t + B - 1) / B), B, 0, stream>>>(agg2, mbuf, dinv, b2, W3, z, acc, N);
    }

    // 8) layer-3 edge reduction
    l3_reduce<<<1024, B, 0, stream>>>(row, norm, z, acc, E);

    // 9) sigmoid(acc + N*b3)
    finalize_kernel<<<1, 1, 0, stream>>>(acc, b3, out, N);
}